// MaxProdRelOut_25374666784830
// MI455X (gfx1250) — compile-verified
//
#include <hip/hip_runtime.h>
#include <cstdint>

#define EPSF 1e-07f

typedef __attribute__((ext_vector_type(2))) float v2f;
typedef __attribute__((ext_vector_type(8))) float v8f;
typedef __attribute__((ext_vector_type(4))) int v4i;

#define AS1 __attribute__((address_space(1)))
#define AS3 __attribute__((address_space(3)))

// ---------------------------------------------------------------------------
// gfx1250 async global->LDS copy (16B per lane). The toolchain's builtin takes
// int4-typed address-space pointers: (v4i as1*, v4i as3*, imm offset, imm cpol).
// Falls back to a plain vector copy if the builtin is not declared.
// ---------------------------------------------------------------------------
__device__ __forceinline__ void cp_async_f4(float* lds_dst, const float* g_src) {
#if __has_builtin(__builtin_amdgcn_global_load_async_to_lds_b128)
  __builtin_amdgcn_global_load_async_to_lds_b128(
      (AS1 v4i*)(uintptr_t)g_src,
      (AS3 v4i*)(uint32_t)(uintptr_t)lds_dst,
      0, 0);
#else
  *(float4*)lds_dst = *(const float4*)g_src;
#endif
}

__device__ __forceinline__ void cp_async_wait_all() {
#if __has_builtin(__builtin_amdgcn_global_load_async_to_lds_b128)
#if __has_builtin(__builtin_amdgcn_s_wait_asynccnt)
  __builtin_amdgcn_s_wait_asynccnt(0);
#else
  asm volatile("s_wait_asynccnt 0" ::: "memory");
#endif
#endif
  __syncthreads();
}

// ---------------------------------------------------------------------------
// Kernel 1: reciprocals  r = 1/(x+eps) (256x256),  q = 1/(w+eps) (256x512)
// Hoists 67M divides in the reference down to 196K.
// ---------------------------------------------------------------------------
__global__ __launch_bounds__(256) void recip_kernel(const float* __restrict__ x,
                                                    const float* __restrict__ w,
                                                    float* __restrict__ r,
                                                    float* __restrict__ q) {
  int idx = blockIdx.x * 256 + threadIdx.x;
  if (idx < 65536) {
    r[idx] = 1.0f / (x[idx] + EPSF);
  } else if (idx < 196608) {
    int j = idx - 65536;
    q[j] = 1.0f / (w[j] + EPSF);
  }
}

// ---------------------------------------------------------------------------
// Kernel 2: row sums and column sums of x via V_WMMA_F32_16X16X4_F32.
// A = 16x4 all-ones  =>  D[m,n] = sum_k B[k,n]  (every row of D equal).
// Waves 0..15: column sums (sum over b). Waves 16..31: row sums (sum over i).
// Wave-uniform branches keep EXEC all-ones around WMMA.
// ---------------------------------------------------------------------------
__global__ __launch_bounds__(1024) void sums_kernel(const float* __restrict__ x,
                                                    float* __restrict__ s_col,
                                                    float* __restrict__ s_row) {
  const int tid  = threadIdx.x;
  const int wave = tid >> 5;
  const int lane = tid & 31;
#if __has_builtin(__builtin_amdgcn_wmma_f32_16x16x4_f32)
  const int n  = lane & 15;
  const int hi = lane >> 4;
  v2f a;
  a.x = 1.0f;
  a.y = 1.0f;
  v8f acc = {};
  if (wave < 16) {
    const int c0 = wave * 16;  // columns [c0, c0+16)
#pragma unroll 4
    for (int b0 = 0; b0 < 256; b0 += 4) {
      v2f bv;
      bv.x = x[(b0 + 2 * hi + 0) * 256 + c0 + n];
      bv.y = x[(b0 + 2 * hi + 1) * 256 + c0 + n];
      acc = __builtin_amdgcn_wmma_f32_16x16x4_f32(false, a, false, bv,
                                                  (short)0, acc, false, false);
    }
    if (lane < 16) s_col[c0 + lane] = acc[0];
  } else {
    const int r0 = (wave - 16) * 16;  // rows [r0, r0+16)
#pragma unroll 4
    for (int i0 = 0; i0 < 256; i0 += 4) {
      v2f bv;
      bv.x = x[(r0 + n) * 256 + i0 + 2 * hi + 0];
      bv.y = x[(r0 + n) * 256 + i0 + 2 * hi + 1];
      acc = __builtin_amdgcn_wmma_f32_16x16x4_f32(false, a, false, bv,
                                                  (short)0, acc, false, false);
    }
    if (lane < 16) s_row[r0 + lane] = acc[0];
  }
#else
  // Scalar fallback.
  if (tid < 256) {
    float s = 0.f;
    for (int b = 0; b < 256; ++b) s += x[b * 256 + tid];
    s_col[tid] = s;
  } else if (tid < 512) {
    int b = tid - 256;
    float s = 0.f;
    for (int i = 0; i < 256; ++i) s += x[b * 256 + i];
    s_row[b] = s;
  }
#endif
}

// ---------------------------------------------------------------------------
// Kernel 3: rel_x[i,o] = (sum_b min(t[b,o]*r[b,i], 1)) / s_col[i]
// Tile: 32 i x 16 o per block; stage t (256x16, 16KB) and r (256x32, 32KB)
// into LDS with async copies; each thread owns (2 i, 1 o).
// ---------------------------------------------------------------------------
#define K3_ITILE 32
#define K3_OTILE 16
__global__ __launch_bounds__(256) void relx_kernel(const float* __restrict__ t,
                                                   const float* __restrict__ r,
                                                   const float* __restrict__ s_col,
                                                   float* __restrict__ rel_x) {
  __shared__ __attribute__((aligned(16))) float t_tile[256 * K3_OTILE];  // [b][o]
  __shared__ __attribute__((aligned(16))) float r_tile[256 * K3_ITILE];  // [b][i]
  const int i0 = blockIdx.x * K3_ITILE;
  const int o0 = blockIdx.y * K3_OTILE;
  const int tid = threadIdx.x;

  for (int c = tid; c < 256 * (K3_OTILE / 4); c += 256) {
    int b = c / (K3_OTILE / 4), p = c % (K3_OTILE / 4);
    cp_async_f4(&t_tile[b * K3_OTILE + p * 4], t + b * 512 + o0 + p * 4);
  }
  for (int c = tid; c < 256 * (K3_ITILE / 4); c += 256) {
    int b = c / (K3_ITILE / 4), p = c % (K3_ITILE / 4);
    cp_async_f4(&r_tile[b * K3_ITILE + p * 4], r + b * 256 + i0 + p * 4);
  }
  cp_async_wait_all();

  const int o_loc = tid & 15;
  const int i_loc = (tid >> 4) * 2;
  float acc0 = 0.f, acc1 = 0.f;
#pragma unroll 4
  for (int b = 0; b < 256; ++b) {
    float tv  = t_tile[b * K3_OTILE + o_loc];
    float r0v = r_tile[b * K3_ITILE + i_loc];
    float r1v = r_tile[b * K3_ITILE + i_loc + 1];
    acc0 += fminf(tv * r0v, 1.0f);
    acc1 += fminf(tv * r1v, 1.0f);
  }
  rel_x[(i0 + i_loc + 0) * 512 + o0 + o_loc] = acc0 / s_col[i0 + i_loc + 0];
  rel_x[(i0 + i_loc + 1) * 512 + o0 + o_loc] = acc1 / s_col[i0 + i_loc + 1];
}

// ---------------------------------------------------------------------------
// Kernel 4: rel_w[b,o] = (sum_i min(t[b,o]*q[i,o], 1)) / s_row[b]
// Tile: 16 b x 16 o per block; q tile (256x16, 16KB) in LDS, reused by all b.
// ---------------------------------------------------------------------------
__global__ __launch_bounds__(256) void relw_kernel(const float* __restrict__ t,
                                                   const float* __restrict__ q,
                                                   const float* __restrict__ s_row,
                                                   float* __restrict__ rel_w) {
  __shared__ __attribute__((aligned(16))) float q_tile[256 * 16];  // [i][o]
  const int o0 = blockIdx.x * 16;
  const int b0 = blockIdx.y * 16;
  const int tid = threadIdx.x;

  for (int c = tid; c < 1024; c += 256) {
    int i = c >> 2, p = c & 3;
    cp_async_f4(&q_tile[i * 16 + p * 4], q + i * 512 + o0 + p * 4);
  }
  cp_async_wait_all();

  const int o_loc = tid & 15, b_loc = tid >> 4;
  const float s = t[(b0 + b_loc) * 512 + o0 + o_loc];
  float acc = 0.f;
#pragma unroll 8
  for (int i = 0; i < 256; ++i) acc += fminf(s * q_tile[i * 16 + o_loc], 1.0f);
  rel_w[(b0 + b_loc) * 512 + o0 + o_loc] = acc / s_row[b0 + b_loc];
}

// ---------------------------------------------------------------------------
// Kernel 5: ind_x = argmax_i x[b,i]*rel_x[i,o] (first-index ties, matching
// jnp.argmax), then chosen_x[b,o] = x[b,ind]*w[ind,o].
// ---------------------------------------------------------------------------
__global__ __launch_bounds__(256) void chosenx_kernel(const float* __restrict__ x,
                                                      const float* __restrict__ w,
                                                      const float* __restrict__ rel_x,
                                                      float* __restrict__ out) {
  __shared__ __attribute__((aligned(16))) float rx_tile[256 * 16];  // [i][o]
  __shared__ __attribute__((aligned(16))) float x_tile[16 * 256];   // [b][i]
  const int b0 = blockIdx.x * 16;
  const int o0 = blockIdx.y * 16;
  const int tid = threadIdx.x;

  for (int c = tid; c < 1024; c += 256) {
    int i = c >> 2, p = c & 3;
    cp_async_f4(&rx_tile[i * 16 + p * 4], rel_x + i * 512 + o0 + p * 4);
  }
  for (int c = tid; c < 1024; c += 256) {
    int bl = c >> 6, p = c & 63;
    cp_async_f4(&x_tile[bl * 256 + p * 4], x + (b0 + bl) * 256 + p * 4);
  }
  cp_async_wait_all();

  const int o_loc = tid & 15, b_loc = tid >> 4;
  float best = -__builtin_inff();
  int bi = 0;
#pragma unroll 4
  for (int i = 0; i < 256; ++i) {
    float v = x_tile[b_loc * 256 + i] * rx_tile[i * 16 + o_loc];
    if (v > best) { best = v; bi = i; }
  }
  out[(b0 + b_loc) * 512 + o0 + o_loc] =
      x_tile[b_loc * 256 + bi] * w[bi * 512 + o0 + o_loc];
}

// ---------------------------------------------------------------------------
// Kernel 6: ind_w = argmax_i rel_w[b,o]*w[i,o], chosen_w[b,o] = x[b,ind]*w[ind,o].
// ---------------------------------------------------------------------------
__global__ __launch_bounds__(256) void chosenw_kernel(const float* __restrict__ x,
                                                      const float* __restrict__ w,
                                                      const float* __restrict__ rel_w,
                                                      float* __restrict__ out) {
  __shared__ __attribute__((aligned(16))) float w_tile[256 * 16];  // [i][o]
  const int b0 = blockIdx.x * 16;
  const int o0 = blockIdx.y * 16;
  const int tid = threadIdx.x;

  for (int c = tid; c < 1024; c += 256) {
    int i = c >> 2, p = c & 3;
    cp_async_f4(&w_tile[i * 16 + p * 4], w + i * 512 + o0 + p * 4);
  }
  cp_async_wait_all();

  const int o_loc = tid & 15, b_loc = tid >> 4;
  const float s = rel_w[(b0 + b_loc) * 512 + o0 + o_loc];
  float best = -__builtin_inff();
  int bi = 0;
  float bw = 0.f;
#pragma unroll 4
  for (int i = 0; i < 256; ++i) {
    float wv = w_tile[i * 16 + o_loc];
    float v = s * wv;
    if (v > best) { best = v; bi = i; bw = wv; }
  }
  out[(b0 + b_loc) * 512 + o0 + o_loc] = x[(b0 + b_loc) * 256 + bi] * bw;
}

// ---------------------------------------------------------------------------
// Host launcher. Inputs: x (256x256), w (256x512), t (256x512), all fp32.
// Output: chosen_x (256x512) then chosen_w (256x512), concatenated.
// Workspace (floats): r 65536 | q 131072 | s_col 256 | s_row 256 |
//                     rel_x 131072 | rel_w 131072   (total ~1.75 MB)
// ---------------------------------------------------------------------------
extern "C" void kernel_launch(void* const* d_in, const int* in_sizes, int n_in,
                              void* d_out, int out_size, void* d_ws, size_t ws_size,
                              hipStream_t stream) {
  const float* x = (const float*)d_in[0];
  const float* w = (const float*)d_in[1];
  const float* t = (const float*)d_in[2];
  float* out = (float*)d_out;
  float* ws = (float*)d_ws;

  float* r     = ws;            // 65536
  float* q     = ws + 65536;    // 131072
  float* s_col = ws + 196608;   // 256
  float* s_row = ws + 196864;   // 256
  float* rel_x = ws + 197120;   // 131072
  float* rel_w = ws + 328192;   // 131072

  recip_kernel<<<768, 256, 0, stream>>>(x, w, r, q);
  sums_kernel<<<1, 1024, 0, stream>>>(x, s_col, s_row);
  relx_kernel<<<dim3(8, 32), 256, 0, stream>>>(t, r, s_col, rel_x);
  relw_kernel<<<dim3(32, 16), 256, 0, stream>>>(t, q, s_row, rel_w);
  chosenx_kernel<<<dim3(16, 32), 256, 0, stream>>>(x, w, rel_x, out);
  chosenw_kernel<<<dim3(16, 32), 256, 0, stream>>>(x, w, rel_w, out + 131072);
}